// InterpretableTensorClassifier_82154134438096
// MI455X (gfx1250) — compile-verified
//
#include <hip/hip_runtime.h>
#include <hip/hip_bf16.h>

// CDNA5 / gfx1250: wave32, WMMA f32 16x16x4, async global->LDS copies.
typedef __attribute__((ext_vector_type(2))) float v2f;
typedef __attribute__((ext_vector_type(8))) float v8f;

#define D_DIM 32
#define NCOL  1024      // 32*32 columns of each mode unfolding
#define RANK  4
#define HID   32
#define NITER 10

__device__ __forceinline__ float wave_sum(float v) {
  v += __shfl_xor(v, 16, 32);
  v += __shfl_xor(v, 8, 32);
  v += __shfl_xor(v, 4, 32);
  v += __shfl_xor(v, 2, 32);
  v += __shfl_xor(v, 1, 32);
  return v;
}

// LDS flat index of element (row, c) of the mode-MODE unfolding of T[i][j][l]
// (T stored row-major with strides 1024/32/1). MODE is compile-time so the
// WMMA inner loops contain no divergence / EXEC manipulation.
template <int MODE>
__device__ __forceinline__ int unf_idx_t(int row, int c) {
  if (MODE == 0) return row * NCOL + c;                                // c = j*32+l
  else if (MODE == 1) return (c >> 5) * NCOL + row * D_DIM + (c & 31); // c = i*32+l
  else return (c >> 5) * NCOL + (c & 31) * D_DIM + row;                // c = i*32+j
}

// One 16x16 tile (p,q) of G = U_MODE * U_MODE^T, K = 1024 swept 4 at a time.
// A 16x4 f32 fragment: lanes 0-15 hold K=0,1; lanes 16-31 hold K=2,3.
template <int MODE>
__device__ __forceinline__ v8f gram_tile(const float* __restrict__ sT,
                                         int p, int q, int lane) {
  const int m  = lane & 15;
  const int ko = (lane >> 4) << 1;
  const int ra = p * 16 + m;
  const int rb = q * 16 + m;
  v8f acc = {};
#pragma unroll 4
  for (int k = 0; k < NCOL; k += 4) {
    v2f av, bv;
    av.x = sT[unf_idx_t<MODE>(ra, k + ko)];
    av.y = sT[unf_idx_t<MODE>(ra, k + ko + 1)];
    bv.x = sT[unf_idx_t<MODE>(rb, k + ko)];
    bv.y = sT[unf_idx_t<MODE>(rb, k + ko + 1)];
    acc = __builtin_amdgcn_wmma_f32_16x16x4_f32(false, av, false, bv,
                                                (short)0, acc, false, false);
  }
  return acc;
}

// 16-row tile p of M = U_MODE (32x1024) @ Krp (1024x4), K range [k0, k0+512).
// Krp occupies B-fragment columns N=0..3; each wave reduces half of K so all
// 4 waves of the block participate (partials summed later).
template <int MODE>
__device__ __forceinline__ v8f mttkrp_tile(const float* __restrict__ sT,
                                           const float* __restrict__ sKrp,
                                           int p, int k0, int lane) {
  const int m  = lane & 15;
  const int ko = (lane >> 4) << 1;
  const int ra = p * 16 + m;
  v8f acc = {};
#pragma unroll 4
  for (int k = k0; k < k0 + (NCOL / 2); k += 4) {
    v2f av, bv;
    av.x = sT[unf_idx_t<MODE>(ra, k + ko)];
    av.y = sT[unf_idx_t<MODE>(ra, k + ko + 1)];
    bv.x = (m < RANK) ? sKrp[(k + ko) * 4 + m] : 0.f;
    bv.y = (m < RANK) ? sKrp[(k + ko + 1) * 4 + m] : 0.f;
    acc = __builtin_amdgcn_wmma_f32_16x16x4_f32(false, av, false, bv,
                                                (short)0, acc, false, false);
  }
  return acc;
}

// Modified Gram-Schmidt on 4 per-lane column values (32 rows = 32 lanes).
__device__ __forceinline__ void mgs4(float x[RANK]) {
#pragma unroll
  for (int r = 0; r < RANK; ++r) {
    float nrm = wave_sum(x[r] * x[r]);
    float inv = rsqrtf(nrm + 1e-20f);
    x[r] *= inv;
#pragma unroll
    for (int s = r + 1; s < RANK; ++s) {
      float d = wave_sum(x[s] * x[r]);
      x[s] -= d * x[r];
    }
  }
}

__global__ __launch_bounds__(128)
void cp_als_fused_kernel(
    const float* __restrict__ T,
    const float* __restrict__ seed_ic,  const float* __restrict__ seed_sir,
    const float* __restrict__ seed_icc, const float* __restrict__ seed_sirc,
    const float* __restrict__ scalars,
    const float* __restrict__ cp_W1,   const float* __restrict__ cp_b1,
    const float* __restrict__ cp_W2,   const float* __restrict__ cp_b2,
    const float* __restrict__ seed_W1, const float* __restrict__ seed_b1,
    const float* __restrict__ seed_W2, const float* __restrict__ seed_b2,
    const float* __restrict__ fin_W1,  const float* __restrict__ fin_b1,
    const float* __restrict__ fin_W2,  const float* __restrict__ fin_b2,
    float* __restrict__ out) {
  extern __shared__ float smem[];
  float* sT    = smem;           // 32768 : the whole 32^3 sample tensor
  float* sG    = sT + 32768;     // 3072  : three 32x32 Gram matrices
  float* sF    = sG + 3072;      // 384   : three 32x4 factor matrices
  float* sKrp  = sF + 384;       // 4096  : 1024x4 Khatri-Rao product
  float* sM    = sKrp + 4096;    // 128   : 32x4 MTTKRP partial (K lower half)
  float* sM2   = sM + 128;       // 128   : 32x4 MTTKRP partial (K upper half)
  float* sV    = sM2 + 128;      // 16    : 4x4 normal matrix
  float* sTrip = sV + 16;        // 16    : [R,3] factor means
  float* sComb = sTrip + 16;     // 16    : final 12-vector

  const int b    = blockIdx.x;
  const int t    = threadIdx.x;
  const int lane = t & 31;
  const int wave = t >> 5;

  // ---- Stage 0: async DMA of the sample tensor into LDS (ASYNCcnt path) ----
  {
    const float* gsrc = T + (size_t)b * 32768;
    unsigned lbase = (unsigned)(unsigned long long)sT;  // flat low 32 bits == LDS byte addr
    for (int c = 0; c < 64; ++c) {
      int e = (c * 128 + t) * 4;                         // 4 floats / b128 chunk
      unsigned laddr = lbase + (unsigned)(e * 4);
      const float* g = gsrc + e;
      asm volatile("global_load_async_to_lds_b128 %0, %1, off"
                   :: "v"(laddr), "v"(g) : "memory");
    }
    asm volatile("s_wait_asynccnt 0" ::: "memory");
  }
  __syncthreads();

  // ---- Stage 1: Gram matrices G_m = U_m U_m^T via V_WMMA_F32_16X16X4_F32 ----
  // 12 tile jobs (3 modes x 2x2 tiles of 16x16) spread over 4 waves.
  for (int jj = wave; jj < 12; jj += 4) {
    const int mode = jj >> 2;
    const int p = (jj >> 1) & 1;
    const int q = jj & 1;
    v8f acc;
    switch (mode) {                  // wave-uniform dispatch, no divergence
      case 0:  acc = gram_tile<0>(sT, p, q, lane); break;
      case 1:  acc = gram_tile<1>(sT, p, q, lane); break;
      default: acc = gram_tile<2>(sT, p, q, lane); break;
    }
    const int row0 = p * 16 + (lane >> 4) * 8;   // D layout: vgpr v -> M=v(+8)
    const int col  = q * 16 + (lane & 15);
#pragma unroll
    for (int v = 0; v < 8; ++v)
      sG[mode * 1024 + (row0 + v) * 32 + col] = acc[v];
  }
  __syncthreads();

  // ---- Stage 2: top-4 invariant subspace of each Gram (orthogonal iteration) ----
  if (wave < 3) {
    const int mode = wave;
    float* sGm = sG + mode * 1024;
    float* sFm = sF + mode * 128;
    const int i = lane;          // lane owns row i of the 32x4 factor
    float x[RANK];
#pragma unroll
    for (int r = 0; r < RANK; ++r) x[r] = sGm[i * 32 + r];
    for (int it = 0; it < 8; ++it) {
      mgs4(x);
#pragma unroll
      for (int r = 0; r < RANK; ++r) sFm[i * RANK + r] = x[r];  // in-order DS within wave
      float y[RANK] = {0.f, 0.f, 0.f, 0.f};
      for (int j = 0; j < 32; ++j) {
        float g = sGm[i * 32 + j];
#pragma unroll
        for (int r = 0; r < RANK; ++r) y[r] += g * sFm[j * RANK + r];
      }
#pragma unroll
      for (int r = 0; r < RANK; ++r) x[r] = y[r];
    }
    mgs4(x);
#pragma unroll
    for (int r = 0; r < RANK; ++r) sFm[i * RANK + r] = x[r];
  }
  __syncthreads();

  // ---- Stage 3: 10 ALS sweeps, modes 0/1/2 ----
  for (int it = 0; it < NITER; ++it) {
    for (int mode = 0; mode < 3; ++mode) {
      const float* fa = sF + ((mode == 0) ? 1 : 0) * 128;
      const float* fb = sF + ((mode == 2) ? 1 : 2) * 128;
      // Khatri-Rao product: Krp[c, r] = fa[c>>5, r] * fb[c&31, r].
      // 32 contiguous floats per thread -> ds_store_b128 friendly.
      {
        const int e0 = t * 32;
#pragma unroll
        for (int e = e0; e < e0 + 32; e += 4) {
          const int c = e >> 2;
          const int ia = (c >> 5) * 4, ib = (c & 31) * 4;
#pragma unroll
          for (int r = 0; r < 4; ++r) sKrp[e + r] = fa[ia + r] * fb[ib + r];
        }
      }
      // V = (fa'fa) hadamard (fb'fb)  (16 entries on threads 0..15)
      if (t < 16) {
        int r = t >> 2, s = t & 3;
        float sa = 0.f, sb = 0.f;
        for (int j = 0; j < 32; ++j) {
          sa += fa[j * 4 + r] * fa[j * 4 + s];
          sb += fb[j * 4 + r] * fb[j * 4 + s];
        }
        sV[t] = sa * sb;
      }
      __syncthreads();
      // MTTKRP: M = U_mode (32x1024) @ Krp (1024x4). All 4 waves:
      // wave = (k-half, row-tile); partials summed in the solve stage.
      {
        const int p  = wave & 1;
        const int kh = wave >> 1;
        v8f acc;
        switch (mode) {
          case 0:  acc = mttkrp_tile<0>(sT, sKrp, p, kh * (NCOL / 2), lane); break;
          case 1:  acc = mttkrp_tile<1>(sT, sKrp, p, kh * (NCOL / 2), lane); break;
          default: acc = mttkrp_tile<2>(sT, sKrp, p, kh * (NCOL / 2), lane); break;
        }
        const int m = lane & 15;
        float* dst = kh ? sM2 : sM;
        if (m < RANK) {
          const int row0 = p * 16 + (lane >> 4) * 8;
#pragma unroll
          for (int v = 0; v < 8; ++v) dst[(row0 + v) * 4 + m] = acc[v];
        }
      }
      __syncthreads();
      // Per-row 4x4 solve: F[i,:] = V^{-1} (sM[i,:] + sM2[i,:])
      if (wave == 0) {
        float Vm[4][4], rhs[4];
#pragma unroll
        for (int r = 0; r < 4; ++r) {
#pragma unroll
          for (int s = 0; s < 4; ++s) Vm[r][s] = sV[r * 4 + s];
          rhs[r] = sM[lane * 4 + r] + sM2[lane * 4 + r];
        }
#pragma unroll
        for (int c = 0; c < 4; ++c) {
          float piv = Vm[c][c];
          piv = (fabsf(piv) > 1e-20f) ? piv : 1e-20f;
          float inv = 1.f / piv;
#pragma unroll
          for (int s = c + 1; s < 4; ++s) {
            float f = Vm[s][c] * inv;
#pragma unroll
            for (int c2 = c; c2 < 4; ++c2) Vm[s][c2] -= f * Vm[c][c2];
            rhs[s] -= f * rhs[c];
          }
        }
#pragma unroll
        for (int c = 3; c >= 0; --c) {
          float v = rhs[c];
#pragma unroll
          for (int c2 = c + 1; c2 < 4; ++c2) v -= Vm[c][c2] * rhs[c2];
          float piv = (fabsf(Vm[c][c]) > 1e-20f) ? Vm[c][c] : 1e-20f;
          rhs[c] = v / piv;
        }
        float* fm = sF + mode * 128;
#pragma unroll
        for (int r = 0; r < 4; ++r) fm[lane * 4 + r] = rhs[r];
      }
      __syncthreads();
    }
  }

  // ---- Stage 4: factor means + MLP heads ----
  if (t < 12) {
    int which = t >> 2, r = t & 3;
    float s = 0.f;
    for (int j = 0; j < 32; ++j) s += sF[which * 128 + j * 4 + r];
    sTrip[r * 3 + which] = s * (1.f / 32.f);
  }
  if (t >= 120 && t < 124) sComb[8 + (t - 120)] = scalars[(size_t)b * 4 + (t - 120)];
  __syncthreads();

  if (wave == 0) {          // 4 CP MLPs: Linear(3,32)->ReLU->Linear(32,1)
    for (int r = 0; r < RANK; ++r) {
      float acc = cp_b1[r * HID + lane];
#pragma unroll
      for (int u = 0; u < 3; ++u)
        acc += cp_W1[(r * HID + lane) * 3 + u] * sTrip[r * 3 + u];
      float h = fmaxf(acc, 0.f);
      float vs = wave_sum(cp_W2[r * HID + lane] * h);
      if (lane == 0) sComb[r] = vs + cp_b2[r];
    }
  } else if (wave == 1) {   // 4 seed MLPs: Linear(64,32)->ReLU->Linear(32,1)
    const float* seeds[4] = {seed_ic, seed_sir, seed_icc, seed_sirc};
    for (int j = 0; j < 4; ++j) {
      const float* sd = seeds[j] + (size_t)b * 64;
      float acc = seed_b1[j * HID + lane];
      for (int k = 0; k < 64; ++k)
        acc += seed_W1[(j * HID + lane) * 64 + k] * sd[k];
      float h = fmaxf(acc, 0.f);
      float vs = wave_sum(seed_W2[j * HID + lane] * h);
      if (lane == 0) sComb[4 + j] = vs + seed_b2[j];
    }
  }
  __syncthreads();

  if (wave == 0) {          // final head: Linear(12,32)->ReLU->Linear(32,1)
    float acc = fin_b1[lane];
#pragma unroll
    for (int c = 0; c < 12; ++c) acc += fin_W1[lane * 12 + c] * sComb[c];
    float h = fmaxf(acc, 0.f);
    float vs = wave_sum(fin_W2[lane] * h);
    if (lane == 0) out[b] = vs + fin_b2[0];
  }
}

extern "C" void kernel_launch(void* const* d_in, const int* in_sizes, int n_in,
                              void* d_out, int out_size, void* d_ws, size_t ws_size,
                              hipStream_t stream) {
  (void)n_in; (void)out_size; (void)d_ws; (void)ws_size;
  const float* T         = (const float*)d_in[0];
  const float* seed_ic   = (const float*)d_in[1];
  const float* seed_sir  = (const float*)d_in[2];
  const float* seed_icc  = (const float*)d_in[3];
  const float* seed_sirc = (const float*)d_in[4];
  const float* scalars   = (const float*)d_in[5];
  const float* cp_W1     = (const float*)d_in[6];
  const float* cp_b1     = (const float*)d_in[7];
  const float* cp_W2     = (const float*)d_in[8];
  const float* cp_b2     = (const float*)d_in[9];
  const float* seed_W1   = (const float*)d_in[10];
  const float* seed_b1   = (const float*)d_in[11];
  const float* seed_W2   = (const float*)d_in[12];
  const float* seed_b2   = (const float*)d_in[13];
  const float* fin_W1    = (const float*)d_in[14];
  const float* fin_b1    = (const float*)d_in[15];
  const float* fin_W2    = (const float*)d_in[16];
  const float* fin_b2    = (const float*)d_in[17];
  float* out = (float*)d_out;

  int B = in_sizes[0] / (32 * 32 * 32);   // 2048
  size_t smem = (size_t)(32768 + 3072 + 384 + 4096 + 128 + 128 + 16 + 16 + 16) * sizeof(float);
  cp_als_fused_kernel<<<dim3(B), dim3(128), smem, stream>>>(
      T, seed_ic, seed_sir, seed_icc, seed_sirc, scalars,
      cp_W1, cp_b1, cp_W2, cp_b2,
      seed_W1, seed_b1, seed_W2, seed_b2,
      fin_W1, fin_b1, fin_W2, fin_b2, out);
}